// ConvCaps_3058016714945
// MI455X (gfx1250) — compile-verified
//
#include <hip/hip_runtime.h>
#include <hip/hip_bf16.h>
#include <math.h>

// ---------------- problem constants (from reference) ----------------
constexpr int NB   = 2;            // batch
constexpr int Bc   = 32;           // B input capsule types
constexpr int Cc   = 32;           // C output capsule types
constexpr int Pp   = 4;            // pose dim
constexpr int PS   = Pp * Pp;      // 16
constexpr int K0   = 3, K1 = 3;
constexpr int OH   = 16, OW = 16;  // (18-3+1)
constexpr int BK   = K0 * K1 * Bc; // 288
constexpr int BB   = NB * OH * OW; // 512
constexpr int CH   = Bc * (PS + 1);// 544 channels of x
constexpr int CPS  = Cc * PS;      // 512
constexpr float EPSF = 1e-8f;
constexpr float LAMF = 1e-3f;
constexpr float HALF_LN_2PI = 0.91893853320467274178f;

typedef float v2f __attribute__((ext_vector_type(2)));
typedef float v8f __attribute__((ext_vector_type(8)));

// ---------------- kernel 1: patch gather (pure reinterpretation) ----------------
// p_in flat index == flat index of p[..., :512] over (2,3,3,16,16,512)
// a_in flat index == flat index of p[..., 512:] over (2,3,3,16,16,32)
__global__ void convcaps_gather(const float* __restrict__ x,
                                float* __restrict__ p_in,
                                float* __restrict__ a_in) {
    const int NP = BB * BK * PS;   // 2,359,296
    const int NA = BB * BK;        // 147,456
    int tid = blockIdx.x * blockDim.x + threadIdx.x;
    if (tid < NP) {
        int f = tid;
        int c  = f % (Bc * PS); f /= (Bc * PS);
        int ww = f % OW;        f /= OW;
        int hh = f % OH;        f /= OH;
        int kj = f % K1;        f /= K1;
        int ki = f % K0;        f /= K0;
        int b  = f;
        p_in[tid] = x[((b * 18 + ki + hh) * 18 + (kj + ww)) * CH + c];
    } else if (tid < NP + NA) {
        int f = tid - NP;
        int g = f;
        int c  = g % Bc; g /= Bc;
        int ww = g % OW; g /= OW;
        int hh = g % OH; g /= OH;
        int kj = g % K1; g /= K1;
        int ki = g % K0; g /= K0;
        int b  = g;
        a_in[f] = x[((b * 18 + ki + hh) * 18 + (kj + ww)) * CH + Bc * PS + c];
    }
}

// ---------------- kernel 2: vote transform via V_WMMA_F32_16X16X4_F32 -------------
// For each n: D(2048x128) = A(2048x4) * B(4x128)
//   A[m][j]        = p_in[m>>2][n][(m&3)*4 + j]     (m = bb*4 + i)
//   B[j][k*4 + l]  = weights[n][k][j][l]
//   v[bb][n][k][i*4+l] = D[bb*4+i][k*4+l]
// One wave per (n, N-tile, M-chunk); B fragment loaded once, 16 WMMAs per wave.
__global__ __launch_bounds__(256)
void convcaps_vote_wmma(const float* __restrict__ p_in,
                        const float* __restrict__ w,
                        float* __restrict__ v) {
    int wave = (blockIdx.x * blockDim.x + threadIdx.x) >> 5;
    int lane = threadIdx.x & 31;
    int mc = wave & 7;             // M chunk (8 chunks of 16 tiles)
    int nt = (wave >> 3) & 7;      // N tile (8)
    int n  = wave >> 6;            // capsule position (288)
    if (n >= BK) return;           // wave-uniform; grid covers exactly
    int half = lane >> 4;          // K-pair select per ISA f32 layouts
    int l16  = lane & 15;

    // B fragment (weights) loaded ONCE per wave:
    // lane<16 -> rows K={0,1}, lane>=16 -> rows K={2,3}; col N = l16
    int col = nt * 16 + l16;
    int k = col >> 2, l = col & 3;
    const float* bp = w + (((size_t)n * Cc + k) * Pp + half * 2) * Pp + l;
    v2f b; b.x = bp[0]; b.y = bp[Pp];

    float* vout = v + (size_t)n * CPS + k * PS + l;

#pragma unroll 4
    for (int mt = mc * 16; mt < mc * 16 + 16; ++mt) {
        // A fragment: lane<16 -> K={0,1}, lane>=16 -> K={2,3}; row M = l16
        int m  = mt * 16 + l16;
        int bb = m >> 2, i = m & 3;
        const float* ap = p_in + ((size_t)bb * BK + n) * PS + i * Pp + half * 2;
        v2f a; a.x = ap[0]; a.y = ap[1];

        v8f c = {};
        c = __builtin_amdgcn_wmma_f32_16x16x4_f32(false, a, false, b,
                                                  (short)0, c, false, false);

        // D layout: VGPR r -> M = mt*16 + r (+8 for lanes 16..31), N = col
#pragma unroll
        for (int r = 0; r < 8; ++r) {
            int M = mt * 16 + r + half * 8;
            int obb = M >> 2, oi = M & 3;
            vout[(size_t)obb * BK * CPS + oi * Pp] = c[r];
        }
    }
}

// ---------------- kernel 3: fused EM routing (one block per bb) -------------------
// 512 threads: t = k*16 + p. Every v access across the block is one coalesced
// 512-float line per n. r lives in LDS (36 KB). m-step uses the one-pass
// moment identity: sigma = Svv - (2 - s0)*mu^2 + eps, with s0 = rs/(rs+eps).
__global__ __launch_bounds__(512)
void convcaps_routing(const float* __restrict__ v,
                      const float* __restrict__ a_in,
                      const float* __restrict__ beta_u,
                      const float* __restrict__ beta_a,
                      float* __restrict__ out) {
    __shared__ float rbuf[BK * Cc];   // r, then ln_ap, then softmaxed r
    __shared__ float aLDS[BK];
    __shared__ float wn[BK];          // a_in[n] / (a_in[n]*sum_k r + eps)
    __shared__ float rsumS[Cc];
    __shared__ float costS[Cc];
    __shared__ float aoutS[Cc];
    __shared__ float scal[2];         // cmean, cstdv

    const int bb = blockIdx.x;
    const int t  = threadIdx.x;
    const int k  = t >> 4;
    const int p  = t & 15;
    const float* vb = v + (size_t)bb * BK * CPS;

    if (t < BK) aLDS[t] = a_in[(size_t)bb * BK + t];
    for (int idx = t; idx < BK * Cc; idx += 512) rbuf[idx] = 1.0f / Cc;
    __syncthreads();

    float mu_t = 0.0f;
    for (int it = 0; it < 3; ++it) {
        // ---- m-step scalars ----
        if (t < BK) {
            float s = 0.0f;
            for (int kk = 0; kk < Cc; ++kk) s += rbuf[t * Cc + kk];
            float av = aLDS[t];
            wn[t] = av / (av * s + EPSF);
        }
        __syncthreads();
        if (t < Cc) {
            float s = 0.0f;
            for (int n = 0; n < BK; ++n) s += rbuf[n * Cc + t] * wn[n];
            rsumS[t] = s;
        }
        __syncthreads();
        const float rs     = rsumS[k];
        const float inv_rs = 1.0f / (rs + EPSF);
        const float s0     = rs * inv_rs;

        // ---- m-step: single coalesced pass over v ----
        float Sv = 0.0f, Svv = 0.0f;
        for (int n = 0; n < BK; ++n) {
            if (n + 4 < BK) __builtin_prefetch(vb + (n + 4) * CPS + t, 0, 1);
            float coeff = rbuf[n * Cc + k] * wn[n] * inv_rs;
            float vv = vb[n * CPS + t];
            Sv  += coeff * vv;
            Svv += coeff * vv * vv;
        }
        mu_t = Sv;
        float sig_t = Svv - (2.0f - s0) * mu_t * mu_t + EPSF;

        // ---- cost / activation ----
        float ch = (beta_u[t] + 0.5f * __logf(sig_t)) * rs;
        for (int m = 8; m >= 1; m >>= 1) ch += __shfl_xor(ch, m, 32);
        if (p == 0) costS[k] = ch;
        __syncthreads();
        if (t == 0) {
            float s = 0.0f;
            for (int kk = 0; kk < Cc; ++kk) s += costS[kk];
            float cmean = s / Cc;
            float dv = 0.0f;                       // faithful: sum dev, then square
            for (int kk = 0; kk < Cc; ++kk) dv += costS[kk] - cmean;
            scal[0] = cmean;
            scal[1] = sqrtf(dv * dv / Cc + EPSF);
        }
        __syncthreads();
        if (t < Cc) {
            float z = LAMF * (beta_a[t] - (scal[0] - costS[t]) / (scal[1] + EPSF));
            aoutS[t] = 1.0f / (1.0f + __expf(-z));
        }
        __syncthreads();

        // ---- e-step (second coalesced pass over v) ----
        if (it < 2) {
            const float inv2s = 1.0f / (2.0f * sig_t + EPSF);
            const float lgs   = __logf(sqrtf(sig_t) + EPSF);
            const float lga   = __logf(EPSF + aoutS[k]);
            for (int n = 0; n < BK; ++n) {
                if (n + 4 < BK) __builtin_prefetch(vb + (n + 4) * CPS + t, 0, 1);
                float d = vb[n * CPS + t] - mu_t;
                float term = -d * d * inv2s - lgs - HALF_LN_2PI;
                for (int m = 8; m >= 1; m >>= 1) term += __shfl_xor(term, m, 32);
                if (p == 0) rbuf[n * Cc + k] = term + lga;
            }
            __syncthreads();
            // softmax over k for each n
            if (t < BK) {
                float mx = -1e30f;
                for (int kk = 0; kk < Cc; ++kk) mx = fmaxf(mx, rbuf[t * Cc + kk]);
                float s = 0.0f;
                for (int kk = 0; kk < Cc; ++kk) {
                    float e = __expf(rbuf[t * Cc + kk] - mx);
                    rbuf[t * Cc + kk] = e;
                    s += e;
                }
                float inv = 1.0f / s;
                for (int kk = 0; kk < Cc; ++kk) rbuf[t * Cc + kk] *= inv;
            }
            __syncthreads();
        }
    }

    // ---- output: (b,oh,ow, C*PS | C) with bb == (b,oh,ow) row-major ----
    out[(size_t)bb * (CPS + Cc) + t] = mu_t;
    if (t < Cc) out[(size_t)bb * (CPS + Cc) + CPS + t] = aoutS[t];
}

// ---------------- host launcher ----------------
extern "C" void kernel_launch(void* const* d_in, const int* in_sizes, int n_in,
                              void* d_out, int out_size, void* d_ws, size_t ws_size,
                              hipStream_t stream) {
    const float* x      = (const float*)d_in[0];  // (2,18,18,544)
    const float* w      = (const float*)d_in[1];  // (1,288,32,4,4)
    const float* beta_u = (const float*)d_in[2];  // (32,16)
    const float* beta_a = (const float*)d_in[3];  // (32,)
    float* out = (float*)d_out;                   // (2,16,16,544)

    // workspace: p_in (9.4MB) | a_in (0.6MB) | v (302MB)
    float* p_in = (float*)d_ws;
    float* a_ws = p_in + (size_t)BB * BK * PS;
    float* v    = a_ws + (size_t)BB * BK;

    {   // gather
        int total = BB * BK * PS + BB * BK;
        convcaps_gather<<<(total + 255) / 256, 256, 0, stream>>>(x, p_in, a_ws);
    }
    {   // votes: 288 n * 8 Ntiles * 8 Mchunks waves, 8 waves/block, 16 WMMA each
        int waves  = BK * 8 * 8;   // 18432
        convcaps_vote_wmma<<<waves / 8, 256, 0, stream>>>(p_in, w, v);
    }
    {   // routing: one block per output position
        convcaps_routing<<<BB, 512, 0, stream>>>(v, a_ws, beta_u, beta_a, out);
    }
}